// PairWiseInterationLayer_17463337025632
// MI455X (gfx1250) — compile-verified
//
#include <hip/hip_runtime.h>
#include <stdint.h>

// Pairwise interaction layer:  out[b, p, :] = in[b, i(p), :] * in[b, j(p), :]
// B=4096, F=40, E=64, P = F*(F-1)/2 = 780.
// Roofline: 0.2 GFLOP vs ~860 MB traffic (817 MB write-once output stream)
// -> pure bandwidth problem, floor ~37 us at 23.3 TB/s.
//
// MI455X strategy:
//  - 1 block (256 threads = 8 wave32) per batch element.
//  - Stage the contiguous 10 KB fp32 tile into LDS with ONE Tensor Data Mover
//    op (tensor_load_to_lds, TENSORcnt), issued by wave 0, drained with
//    s_wait_tensorcnt, published to the block with s_barrier.
//  - Hot loop: 2x ds_load_b128 + 1x non-temporal global_store_b128 per 16 B
//    of output; NT keeps the 192 MB L2 from being thrashed by the 817 MB
//    write-once stream.

typedef float        v4f  __attribute__((ext_vector_type(4)));
typedef unsigned int u32x4 __attribute__((ext_vector_type(4)));
typedef unsigned int u32x8 __attribute__((ext_vector_type(8)));

#define F_DIM   40
#define E_DIM   64
#define NPAIR   (F_DIM * (F_DIM - 1) / 2)   // 780
#define TILE_N  (F_DIM * E_DIM)             // 2560 floats = 10 KB
#define OUT_PER_B (NPAIR * E_DIM)           // 49920 floats
#define THREADS 256

__global__ __launch_bounds__(THREADS)
void pairwise_interaction_kernel(const float* __restrict__ in,
                                 float* __restrict__ out) {
    __shared__ __align__(16) float tile[TILE_N];
    __shared__ unsigned short pairTab[NPAIR];

    const int tid = threadIdx.x;
    const int b   = blockIdx.x;

    // ---- p -> (i<<8)|j table, closed form (i<j, row-major (i,j) order) ----
    // cum(i) = i*(79-i)/2 pairs precede row i (F=40).
    for (int p = tid; p < NPAIR; p += THREADS) {
        int i = (int)((79.0f - sqrtf(6241.0f - 8.0f * (float)p)) * 0.5f);
        while ((i + 1) * (79 - (i + 1)) / 2 <= p) ++i;   // exact fixup
        while (i * (79 - i) / 2 > p) --i;
        const int j = p - i * (79 - i) / 2 + i + 1;
        pairTab[p] = (unsigned short)((i << 8) | j);
    }

    // ---- TDM: one tensor_load_to_lds for the whole 10 KB tile ----
    if (tid == 0) {
        const uint64_t ga  = (uint64_t)(uintptr_t)(in + (size_t)b * TILE_N);
        const unsigned lds = (unsigned)(uintptr_t)(void*)tile; // low 32b of generic = LDS offset

        u32x4 g0;
        g0[0] = 1u;                                    // count=1, user descriptor
        g0[1] = lds;                                   // lds_addr
        g0[2] = (unsigned)ga;                          // global_addr[31:0]
        g0[3] = ((unsigned)(ga >> 32) & 0x01FFFFFFu)   // global_addr[56:32]
                | (2u << 30);                          // type = 2 ("image")

        u32x8 g1;
        g1[0] = 0x00020000u;                           // workgroup_mask=0, data_size=2 (4 B)
        g1[1] = (TILE_N & 0xFFFFu) << 16;              // tensor_dim0[15:0]  @ bits 63:48
        g1[2] = (TILE_N >> 16) | (1u << 16);           // tensor_dim0[31:16] | tensor_dim1=1
        g1[3] = (TILE_N & 0xFFFFu) << 16;              // tile_dim0 = 2560   @ bits 127:112
        g1[4] = 0u;                                    // tile_dim1=0 (1-D), tile_dim2=0
        g1[5] = (unsigned)TILE_N;                      // tensor_dim0_stride[31:0]
        g1[6] = 0u;                                    // stride hi / dim1_stride lo
        g1[7] = 0u;

        asm volatile("tensor_load_to_lds %0, %1"
                     :
                     : "s"(g0), "s"(g1)
                     : "memory");
        __builtin_amdgcn_s_wait_tensorcnt(0);
    }
    __syncthreads();

    // ---- Hot loop: fixed float4 chunk per thread, pairs strided by 16 ----
    const int    c4    = (tid & 15) * 4;               // float4 chunk in E=64 row
    const float* trow  = tile + c4;
    float*       obase = out + (size_t)b * OUT_PER_B + c4;

    for (int p = tid >> 4; p < NPAIR; p += THREADS / 16) {
        const unsigned pk = pairTab[p];                // broadcast across 16 lanes
        v4f a = *(const v4f*)(trow + (pk >> 8)   * E_DIM);
        v4f w = *(const v4f*)(trow + (pk & 0xFF) * E_DIM);
        __builtin_nontemporal_store(a * w, (v4f*)(obase + (size_t)p * E_DIM));
    }
}

extern "C" void kernel_launch(void* const* d_in, const int* in_sizes, int n_in,
                              void* d_out, int out_size, void* d_ws, size_t ws_size,
                              hipStream_t stream) {
    const float* in  = (const float*)d_in[0];
    float*       out = (float*)d_out;
    const int B = in_sizes[0] / TILE_N;   // 4096 for the reference shapes
    pairwise_interaction_kernel<<<B, THREADS, 0, stream>>>(in, out);
}